// Mamba4Rec_40072044872131
// MI455X (gfx1250) — compile-verified
//
#include <hip/hip_runtime.h>
#include <hip/hip_bf16.h>

#define NL 2
#define D_MODEL 200
#define D_STATE 16
#define D_CONV 4
#define DI 400
#define DT_RANK 13
#define BB 256
#define LL 256
#define M_ROWS (BB * LL) /* 65536 rows, always a multiple of 16 */

typedef __attribute__((ext_vector_type(2))) float v2f;
typedef __attribute__((ext_vector_type(8))) float v8f;

__device__ __forceinline__ float silu_f(float x) { return x / (1.f + expf(-x)); }

// ---------------------------------------------------------------------------
// f32 WMMA GEMM, fully compile-time shaped:
//   C[M,N] = act( A[M,K] @ W[N,K]^T + bias )
// One wave computes one 16x16 tile via V_WMMA_F32_16X16X4_F32 (K step 4).
// A frag: lane = M (lane&15), VGPR v holds K = 2*(lane>>4)+v.
// B frag: lane = N (lane&15), VGPR v holds K = 2*(lane>>4)+v.
// D:      vgpr r, lane -> (M = r + 8*(lane>>4), N = lane&15).
// Main K body: unguarded 8B loads (LDA even, ka even). K tail (K=13 case):
// clamped-index unconditional loads + v_cndmask zeroing — no exec branches.
// ---------------------------------------------------------------------------
#define GEMM_NW 4

template <int LDA, int N, int K, int LDC, int ACT, bool HAS_BIAS>
__global__ __launch_bounds__(32 * GEMM_NW)
void gemm16_wmma(const float* __restrict__ A,
                 const float* __restrict__ W,
                 const float* __restrict__ bias,
                 float* __restrict__ C)
{
  constexpr int NTILES = (N + 15) / 16;
  const int lane = threadIdx.x & 31;
  const int wave = threadIdx.x >> 5;
  const int nt   = blockIdx.y * GEMM_NW + wave;
  if constexpr (NTILES % GEMM_NW != 0) {
    if (nt >= NTILES) return;  // wave-uniform
  }
  const int m0   = blockIdx.x << 4;
  const int n0   = nt << 4;
  const int mrow = m0 + (lane & 15);
  const int kh   = (lane >> 4) << 1;  // 0 or 2
  int wrow = n0 + (lane & 15);
  if (wrow > N - 1) wrow = N - 1;     // clamp (edge tiles; stores guarded)
  const float* Arow = A + (size_t)mrow * LDA;
  const float* Wrow = W + (size_t)wrow * K;

  v8f acc = {0.f, 0.f, 0.f, 0.f, 0.f, 0.f, 0.f, 0.f};
  constexpr int KM = K & ~3;
#pragma unroll 8
  for (int k0 = 0; k0 < KM; k0 += 4) {
    const int ka = k0 + kh;
    const v2f a = *(const v2f*)(Arow + ka);  // 8B aligned: LDA even, ka even
    v2f b;
    if constexpr ((K & 1) == 0) {
      b = *(const v2f*)(Wrow + ka);          // 8B aligned: K even
    } else {
      b.x = Wrow[ka];
      b.y = Wrow[ka + 1];
    }
    acc = __builtin_amdgcn_wmma_f32_16x16x4_f32(false, a, false, b,
                                                (short)0, acc, false, false);
  }
  if constexpr ((K & 3) != 0) {
    // branch-free masked tail: unconditional in-bounds loads, then select 0
    const int ka  = KM + kh;
    const int ia0 = (ka < K) ? ka : (K - 1);
    const int ia1 = (ka + 1 < K) ? (ka + 1) : (K - 1);
    const float a0 = Arow[ia0], a1 = Arow[ia1];
    const float b0 = Wrow[ia0], b1 = Wrow[ia1];
    v2f a, b;
    a.x = (ka < K) ? a0 : 0.f;
    a.y = (ka + 1 < K) ? a1 : 0.f;
    b.x = (ka < K) ? b0 : 0.f;
    b.y = (ka + 1 < K) ? b1 : 0.f;
    acc = __builtin_amdgcn_wmma_f32_16x16x4_f32(false, a, false, b,
                                                (short)0, acc, false, false);
  }

  const int ncol = n0 + (lane & 15);
  if (ncol < N) {
    float bv = 0.f;
    if constexpr (HAS_BIAS) bv = bias[ncol];
    const int mbase = m0 + ((lane >> 4) << 3);
#pragma unroll
    for (int r = 0; r < 8; ++r) {
      float v = acc[r] + bv;
      if constexpr (ACT == 1) v = (v > 20.f) ? v : log1pf(expf(v));  // softplus
      C[(size_t)(mbase + r) * LDC + ncol] = v;
    }
  }
}

// ---------------------------------------------------------------------------
// Depthwise causal conv1d (k=4, left pad 3) + bias + SiLU, in place on the
// xc half of xz[M, 800]. One thread per (batch, channel), register window.
// ---------------------------------------------------------------------------
__global__ __launch_bounds__(256)
void conv_silu_inplace(float* __restrict__ xz,
                       const float* __restrict__ cw,  // [DI,4]
                       const float* __restrict__ cb)  // [DI]
{
  const int idx = blockIdx.x * blockDim.x + threadIdx.x;
  if (idx >= BB * DI) return;
  const int b = idx / DI, c = idx % DI;
  const float w0 = cw[c * 4 + 0], w1 = cw[c * 4 + 1];
  const float w2 = cw[c * 4 + 2], w3 = cw[c * 4 + 3];
  const float bias = cb[c];
  float xm1 = 0.f, xm2 = 0.f, xm3 = 0.f;
  float* p = xz + (size_t)b * LL * (2 * DI) + c;
  for (int l = 0; l < LL; ++l) {
    const float x0 = p[(size_t)l * (2 * DI)];
    const float v  = bias + w3 * x0 + w2 * xm1 + w1 * xm2 + w0 * xm3;
    p[(size_t)l * (2 * DI)] = silu_f(v);
    xm3 = xm2; xm2 = xm1; xm1 = x0;
  }
}

// ---------------------------------------------------------------------------
// Selective scan + output gate, fused.  dy holds delta on entry and the gated
// output yg = (y + u*D) * silu(z) on exit (in-place per element, consumed
// before overwritten).  16 states live in registers; B_t/C_t staged in LDS.
// ---------------------------------------------------------------------------
__global__ __launch_bounds__(128)
void scan_gate(const float* __restrict__ xz,    // [M, 800] (xc | z)
               const float* __restrict__ xdbl,  // [M, 48]  (dt | B | C | pad)
               float* __restrict__ dy,          // [M, 400] delta -> yg
               const float* __restrict__ A_log, // [400,16]
               const float* __restrict__ Dp)    // [400]
{
  __shared__ float sB[D_STATE];
  __shared__ float sC[D_STATE];
  const int d   = blockIdx.x * blockDim.x + threadIdx.x;
  const int b   = blockIdx.y;
  const bool on = (d < DI);

  float Arow[D_STATE], h[D_STATE];
  float Dv = 0.f;
  if (on) {
    Dv = Dp[d];
#pragma unroll
    for (int n = 0; n < D_STATE; ++n) {
      Arow[n] = -expf(A_log[d * D_STATE + n]);
      h[n]    = 0.f;
    }
  }
  for (int l = 0; l < LL; ++l) {
    const size_t row = (size_t)b * LL + l;
    if (threadIdx.x < D_STATE)
      sB[threadIdx.x] = xdbl[row * 48 + DT_RANK + threadIdx.x];
    else if (threadIdx.x < 2 * D_STATE)
      sC[threadIdx.x - D_STATE] = xdbl[row * 48 + DT_RANK + D_STATE + (threadIdx.x - D_STATE)];
    __syncthreads();
    if (on) {
      const float dlt = dy[row * DI + d];
      const float u   = xz[row * (2 * DI) + d];
      const float du  = dlt * u;
      float y = 0.f;
#pragma unroll
      for (int n = 0; n < D_STATE; ++n) {
        const float dA = expf(dlt * Arow[n]);
        h[n] = dA * h[n] + du * sB[n];
        y += h[n] * sC[n];
      }
      const float z = xz[row * (2 * DI) + DI + d];
      dy[row * DI + d] = (y + u * Dv) * silu_f(z);
    }
    __syncthreads();
  }
}

// ---------------------------------------------------------------------------
// h = LayerNorm(m + h) * w + b  — one wave32 per row, shuffle reductions.
// ---------------------------------------------------------------------------
__global__ __launch_bounds__(128)
void add_ln(const float* __restrict__ m, float* __restrict__ h,
            const float* __restrict__ w, const float* __restrict__ bta)
{
  const int lane   = threadIdx.x & 31;
  const int wave   = threadIdx.x >> 5;
  const size_t row = (size_t)blockIdx.x * 4 + wave;
  const float* mp  = m + row * D_MODEL;
  float* hp        = h + row * D_MODEL;

  float vals[7];
  float s = 0.f;
#pragma unroll
  for (int i = 0; i < 7; ++i) {
    const int idx = lane + i * 32;
    float v = 0.f;
    if (idx < D_MODEL) v = mp[idx] + hp[idx];
    vals[i] = v;
    s += v;
  }
#pragma unroll
  for (int off = 16; off > 0; off >>= 1) s += __shfl_xor(s, off, 32);
  const float mu = s * (1.f / D_MODEL);
  float q = 0.f;
#pragma unroll
  for (int i = 0; i < 7; ++i) {
    const int idx = lane + i * 32;
    if (idx < D_MODEL) { const float dd = vals[i] - mu; q += dd * dd; }
  }
#pragma unroll
  for (int off = 16; off > 0; off >>= 1) q += __shfl_xor(q, off, 32);
  const float rstd = rsqrtf(q * (1.f / D_MODEL) + 1e-12f);
#pragma unroll
  for (int i = 0; i < 7; ++i) {
    const int idx = lane + i * 32;
    if (idx < D_MODEL) hp[idx] = (vals[i] - mu) * rstd * w[idx] + bta[idx];
  }
}

// ---------------------------------------------------------------------------
// Fused FFN: per 16-row tile compute f = gelu(h@W1^T + b1) into LDS (stride
// 804 floats -> conflict-free: gcd(36,64)=4, period 16), then Out = f@W2^T+b2
// with WMMA reading the A fragments straight from LDS.
// ---------------------------------------------------------------------------
#define FFN_E 800
#define FFN_LDS 804
#define FFN_NW 8

__global__ __launch_bounds__(32 * FFN_NW)
void ffn_fused(const float* __restrict__ H,
               const float* __restrict__ W1, const float* __restrict__ b1,
               const float* __restrict__ W2, const float* __restrict__ b2,
               float* __restrict__ Out)
{
  __shared__ float f[16 * FFN_LDS];  // 51.4 KB of the 320 KB WGP LDS
  const int lane = threadIdx.x & 31;
  const int wave = threadIdx.x >> 5;
  const int m0   = blockIdx.x << 4;
  const int mrow = m0 + (lane & 15);
  const int kh   = (lane >> 4) << 1;

  // ---- phase 1: f[16,800] = gelu(h @ W1^T + b1) ----
  const float* Arow = H + (size_t)mrow * D_MODEL;
  for (int et = wave; et < FFN_E / 16; et += FFN_NW) {
    const int e0 = et << 4;
    const float* Wr = W1 + (size_t)(e0 + (lane & 15)) * D_MODEL;
    v8f acc = {0.f, 0.f, 0.f, 0.f, 0.f, 0.f, 0.f, 0.f};
#pragma unroll 8
    for (int k0 = 0; k0 < D_MODEL; k0 += 4) {
      const int ka = k0 + kh;
      const v2f a = *(const v2f*)(Arow + ka);
      const v2f b = *(const v2f*)(Wr + ka);
      acc = __builtin_amdgcn_wmma_f32_16x16x4_f32(false, a, false, b,
                                                  (short)0, acc, false, false);
    }
    const int ec = e0 + (lane & 15);
    const float bv = b1[ec];
    const int mb = (lane >> 4) << 3;
#pragma unroll
    for (int r = 0; r < 8; ++r) {
      float v = acc[r] + bv;
      v = 0.5f * v * (1.f + erff(v * 0.70710678118654752f));  // exact gelu
      f[(mb + r) * FFN_LDS + ec] = v;
    }
  }
  __syncthreads();

  // ---- phase 2: Out[16,200] = f @ W2^T + b2 ----
  const float* Fl = f + (lane & 15) * FFN_LDS;
  for (int nt = wave; nt < (D_MODEL + 15) / 16; nt += FFN_NW) {
    const int n0 = nt << 4;
    int wrow = n0 + (lane & 15);
    if (wrow > D_MODEL - 1) wrow = D_MODEL - 1;
    const float* Wr = W2 + (size_t)wrow * FFN_E;
    v8f acc = {0.f, 0.f, 0.f, 0.f, 0.f, 0.f, 0.f, 0.f};
#pragma unroll 8
    for (int k0 = 0; k0 < FFN_E; k0 += 4) {
      const int ka = k0 + kh;
      const v2f a = *(const v2f*)(Fl + ka);   // ds_load_b64 from LDS
      const v2f b = *(const v2f*)(Wr + ka);
      acc = __builtin_amdgcn_wmma_f32_16x16x4_f32(false, a, false, b,
                                                  (short)0, acc, false, false);
    }
    const int nc = n0 + (lane & 15);
    if (nc < D_MODEL) {
      const float bv = b2[nc];
      const int mb = (lane >> 4) << 3;
#pragma unroll
      for (int r = 0; r < 8; ++r)
        Out[(size_t)(m0 + mb + r) * D_MODEL + nc] = acc[r] + bv;
    }
  }
}

// ---------------------------------------------------------------------------
// d_out = concat( h[:, -1, :], h[:, :-1, :] ) flattened in return order.
// ---------------------------------------------------------------------------
__global__ __launch_bounds__(256)
void split_out(const float* __restrict__ h, float* __restrict__ out)
{
  const size_t idx = (size_t)blockIdx.x * blockDim.x + threadIdx.x;
  if (idx >= (size_t)M_ROWS * D_MODEL) return;
  const int dm    = (int)(idx % D_MODEL);
  const size_t rl = idx / D_MODEL;
  const int l = (int)(rl % LL);
  const int b = (int)(rl / LL);
  const float v = h[idx];
  if (l == LL - 1)
    out[(size_t)b * D_MODEL + dm] = v;
  else
    out[(size_t)BB * D_MODEL + ((size_t)b * (LL - 1) + l) * D_MODEL + dm] = v;
}

// ---------------------------------------------------------------------------
extern "C" void kernel_launch(void* const* d_in, const int* in_sizes, int n_in,
                              void* d_out, int out_size, void* d_ws, size_t ws_size,
                              hipStream_t stream)
{
  (void)in_sizes; (void)n_in; (void)out_size; (void)ws_size;
  const float* item_emb   = (const float*)d_in[0];
  const float* in_proj_w  = (const float*)d_in[2];
  const float* conv_w     = (const float*)d_in[3];
  const float* conv_b     = (const float*)d_in[4];
  const float* x_proj_w   = (const float*)d_in[5];
  const float* dt_proj_w  = (const float*)d_in[6];
  const float* dt_proj_b  = (const float*)d_in[7];
  const float* A_log      = (const float*)d_in[8];
  const float* Dp         = (const float*)d_in[9];
  const float* out_proj_w = (const float*)d_in[10];
  const float* ln1_w      = (const float*)d_in[11];
  const float* ln1_b      = (const float*)d_in[12];
  const float* ffn_w1     = (const float*)d_in[13];
  const float* ffn_b1     = (const float*)d_in[14];
  const float* ffn_w2     = (const float*)d_in[15];
  const float* ffn_b2     = (const float*)d_in[16];
  const float* ffn_ln_w   = (const float*)d_in[17];
  const float* ffn_ln_b   = (const float*)d_in[18];

  // workspace layout (floats): ~432 MB total
  float* ws   = (float*)d_ws;
  float* h    = ws;                                   // [M,200]
  float* xz   = h    + (size_t)M_ROWS * D_MODEL;      // [M,800]  (xc | z)
  float* xdbl = xz   + (size_t)M_ROWS * 2 * DI;       // [M,48]   (dt|B|C|pad)
  float* dy   = xdbl + (size_t)M_ROWS * 48;           // [M,400]  delta -> yg
  float* mbuf = dy   + (size_t)M_ROWS * DI;           // [M,200]

  hipMemcpyAsync(h, item_emb, (size_t)M_ROWS * D_MODEL * sizeof(float),
                 hipMemcpyDeviceToDevice, stream);

  const int Mt = M_ROWS / 16;  // 4096 row tiles
  for (int layer = 0; layer < NL; ++layer) {
    // xz = h @ in_proj_w^T                       (N=800 -> 50 tiles, K=200)
    gemm16_wmma<D_MODEL, 2 * DI, D_MODEL, 2 * DI, 0, false>
        <<<dim3(Mt, 13), 32 * GEMM_NW, 0, stream>>>(
            h, in_proj_w + (size_t)layer * 2 * DI * D_MODEL, nullptr, xz);
    // causal depthwise conv + SiLU on xc, in place
    conv_silu_inplace<<<(BB * DI) / 256, 256, 0, stream>>>(
        xz, conv_w + (size_t)layer * DI * D_CONV, conv_b + (size_t)layer * DI);
    // x_dbl = xc @ x_proj_w^T                    (N=45 -> 3 tiles, K=400)
    gemm16_wmma<2 * DI, DT_RANK + 2 * D_STATE, DI, 48, 0, false>
        <<<dim3(Mt, 1), 32 * GEMM_NW, 0, stream>>>(
            xz, x_proj_w + (size_t)layer * (DT_RANK + 2 * D_STATE) * DI,
            nullptr, xdbl);
    // delta = softplus(dt @ dt_proj_w^T + b)     (N=400 -> 25 tiles, K=13)
    gemm16_wmma<48, DI, DT_RANK, DI, 1, true>
        <<<dim3(Mt, 7), 32 * GEMM_NW, 0, stream>>>(
            xdbl, dt_proj_w + (size_t)layer * DI * DT_RANK,
            dt_proj_b + (size_t)layer * DI, dy);
    // selective scan + gate (dy: delta -> yg)
    scan_gate<<<dim3((DI + 127) / 128, BB), 128, 0, stream>>>(
        xz, xdbl, dy, A_log + (size_t)layer * DI * D_STATE, Dp + (size_t)layer * DI);
    // m = yg @ out_proj_w^T                      (N=200 -> 13 tiles, K=400)
    gemm16_wmma<DI, D_MODEL, DI, D_MODEL, 0, false>
        <<<dim3(Mt, 4), 32 * GEMM_NW, 0, stream>>>(
            dy, out_proj_w + (size_t)layer * D_MODEL * DI, nullptr, mbuf);
    // h = LN(m + h)
    add_ln<<<M_ROWS / 4, 128, 0, stream>>>(
        mbuf, h, ln1_w + layer * D_MODEL, ln1_b + layer * D_MODEL);
    // mbuf = FFN(h)  (fused W1+gelu in LDS, then W2)
    ffn_fused<<<Mt, 32 * FFN_NW, 0, stream>>>(
        h, ffn_w1 + (size_t)layer * 4 * D_MODEL * D_MODEL,
        ffn_b1 + (size_t)layer * 4 * D_MODEL,
        ffn_w2 + (size_t)layer * D_MODEL * 4 * D_MODEL,
        ffn_b2 + (size_t)layer * D_MODEL, mbuf);
    // h = LN(mbuf + h)
    add_ln<<<M_ROWS / 4, 128, 0, stream>>>(
        mbuf, h, ffn_ln_w + layer * D_MODEL, ffn_ln_b + layer * D_MODEL);
  }
  split_out<<<(M_ROWS * D_MODEL) / 256, 256, 0, stream>>>(h, (float*)d_out);
}